// DeformConv2d_56014963474890
// MI455X (gfx1250) — compile-verified
//
#include <hip/hip_runtime.h>
#include <cstdint>
#include <cstddef>

#define B_SZ  8
#define CIN   64
#define HIN   128
#define WIN   128
#define COUT  64
#define HW    (HIN * WIN)      // 16384
#define KDIM  (CIN * 9)        // 576

typedef __attribute__((ext_vector_type(2))) float v2f;
typedef __attribute__((ext_vector_type(8))) float v8f;

// ---------------------------------------------------------------------------
// Stage 1: offset-predicting 3x3 conv (pad 1) + bias -> d_ws as [B][18][H][W]
// ---------------------------------------------------------------------------
__global__ __launch_bounds__(256) void deform_offset_conv(
    const float* __restrict__ x, const float* __restrict__ ow,
    const float* __restrict__ ob, float* __restrict__ off)
{
    __shared__ float s_w[18 * 576];
    __shared__ float s_b[18];
    const int tid = threadIdx.x;
    for (int i = tid; i < 18 * 576; i += 256) s_w[i] = ow[i];
    if (tid < 18) s_b[tid] = ob[tid];
    __syncthreads();

    const int n  = blockIdx.x * 256 + tid;
    const int b  = n >> 14;          // / HW
    const int hw = n & (HW - 1);
    const int h  = hw >> 7;
    const int w  = hw & 127;

    float acc[18];
#pragma unroll
    for (int t = 0; t < 18; ++t) acc[t] = s_b[t];

    const float* xb = x + (size_t)b * CIN * HW;
    for (int c = 0; c < CIN; ++c) {
        const float* xc = xb + c * HW;
        float v[9];
#pragma unroll
        for (int ki = 0; ki < 3; ++ki) {
            const int  y  = h - 1 + ki;
            const bool vy = (y >= 0) && (y < HIN);
#pragma unroll
            for (int kj = 0; kj < 3; ++kj) {
                const int  xx = w - 1 + kj;
                const bool vx = (xx >= 0) && (xx < WIN);
                v[ki * 3 + kj] = (vy && vx) ? xc[y * WIN + xx] : 0.0f;
            }
        }
#pragma unroll
        for (int t = 0; t < 18; ++t) {
            const float* wr = &s_w[t * 576 + c * 9];
#pragma unroll
            for (int k = 0; k < 9; ++k) acc[t] = fmaf(v[k], wr[k], acc[t]);
        }
    }

    float* op = off + (size_t)b * 18 * HW + hw;
#pragma unroll
    for (int t = 0; t < 18; ++t) op[(size_t)t * HW] = acc[t];
}

// ---------------------------------------------------------------------------
// Stage 2: fused bilinear sampling + WMMA GEMM
//   out[o, n] = sum_{c,k} W[o, c*9+k] * S[c*9+k, n]
//   M=64 (4 waves x 16), K=576 (144 wmma_f32_16x16x4 steps), N-tile = 16
// ---------------------------------------------------------------------------
__global__ __launch_bounds__(128) void deform_wmma_gemm(
    const float* __restrict__ x, const float* __restrict__ weight,
    const float* __restrict__ off, float* __restrict__ out)
{
    __shared__ float S[KDIM * 16];        // 36 KB sampled tile, S[kidx][pos]
    __shared__ int   pry0[144], pry1[144], prx0[144], prx1[144];
    __shared__ float pwts[144][4];

    const int tid = threadIdx.x;
    const int n0  = blockIdx.x << 4;      // first of 16 positions (same row)
    const int b   = n0 >> 14;
    const int hw0 = n0 & (HW - 1);
    const int h   = hw0 >> 7;
    const int w0  = hw0 & 127;

    // --- 144 (pos, tap) bilinear parameter sets -----------------------------
    if (tid < 144) {
        const int pos = tid / 9;
        const int k   = tid - pos * 9;
        const int w   = w0 + pos;
        const float* ob = off + (size_t)b * 18 * HW + (size_t)h * WIN + w;
        const float dy = ob[(size_t)(k * 2 + 0) * HW];
        const float dx = ob[(size_t)(k * 2 + 1) * HW];
        const float py = (float)(h - 1 + k / 3) + dy;
        const float px = (float)(w - 1 + (k % 3)) + dx;
        const float fy0 = floorf(py), fx0 = floorf(px);
        const float wy1 = py - fy0,   wx1 = px - fx0;
        const float wy0 = 1.0f - wy1, wx0 = 1.0f - wx1;
        const int iy0 = (int)fy0, ix0 = (int)fx0;
        const int iy1 = iy0 + 1,  ix1 = ix0 + 1;
        const float vy0 = (iy0 >= 0 && iy0 < HIN) ? 1.0f : 0.0f;
        const float vy1 = (iy1 >= 0 && iy1 < HIN) ? 1.0f : 0.0f;
        const float vx0 = (ix0 >= 0 && ix0 < WIN) ? 1.0f : 0.0f;
        const float vx1 = (ix1 >= 0 && ix1 < WIN) ? 1.0f : 0.0f;
        const int cy0 = min(max(iy0, 0), HIN - 1);
        const int cy1 = min(max(iy1, 0), HIN - 1);
        const int cx0 = min(max(ix0, 0), WIN - 1);
        const int cx1 = min(max(ix1, 0), WIN - 1);
        pry0[tid] = cy0 * WIN;  pry1[tid] = cy1 * WIN;
        prx0[tid] = cx0;        prx1[tid] = cx1;
        pwts[tid][0] = wy0 * wx0 * vy0 * vx0;
        pwts[tid][1] = wy0 * wx1 * vy0 * vx1;
        pwts[tid][2] = wy1 * wx0 * vy1 * vx0;
        pwts[tid][3] = wy1 * wx1 * vy1 * vx1;
    }
    __syncthreads();

    // --- fill sampled tile S[c*9+k][pos] ------------------------------------
    const float* xb = x + (size_t)b * CIN * HW;
    for (int s = tid; s < KDIM * 16; s += 128) {   // 9216/128 = 72, uniform
        const int pos  = s & 15;
        const int kidx = s >> 4;
        const int c    = kidx / 9;
        const int k    = kidx - c * 9;
        const int pk   = pos * 9 + k;
        const float* xc = xb + c * HW;
        const int r0 = pry0[pk], r1 = pry1[pk];
        const int c0 = prx0[pk], c1 = prx1[pk];
        const float v = pwts[pk][0] * xc[r0 + c0] + pwts[pk][1] * xc[r0 + c1]
                      + pwts[pk][2] * xc[r1 + c0] + pwts[pk][3] * xc[r1 + c1];
        S[kidx * 16 + pos] = v;
    }
    __syncthreads();

    // --- WMMA GEMM: each wave owns 16 output channels -----------------------
    const int wave  = tid >> 5;
    const int lane  = tid & 31;
    const int mbase = wave << 4;
    const int col   = lane & 15;            // N index (and A-row index)
    const int krsel = (lane >> 4) << 1;     // lanes 0-15 -> K 0/1, 16-31 -> K 2/3

    const float* Wr = weight + (size_t)(mbase + col) * KDIM;

    v8f acc = {0.f, 0.f, 0.f, 0.f, 0.f, 0.f, 0.f, 0.f};
#pragma unroll 4
    for (int kb = 0; kb < KDIM / 4; ++kb) {
        const int ka = kb * 4 + krsel;
        v2f a, bmat;
        a.x    = Wr[ka];
        a.y    = Wr[ka + 1];
        bmat.x = S[ka * 16 + col];
        bmat.y = S[(ka + 1) * 16 + col];
        acc = __builtin_amdgcn_wmma_f32_16x16x4_f32(
                  /*neg_a=*/false, a, /*neg_b=*/false, bmat,
                  /*c_mod=*/(short)0, acc, /*reuse_a=*/false, /*reuse_b=*/false);
    }

    // --- scatter D: vgpr i -> M = i (lanes 0-15) / 8+i (lanes 16-31) --------
    const int mrow0 = mbase + ((lane >> 4) << 3);
    float* op = out + (size_t)b * COUT * HW + (size_t)h * WIN + w0 + col;
#pragma unroll
    for (int i = 0; i < 8; ++i)
        op[(size_t)(mrow0 + i) * HW] = acc[i];
}

// ---------------------------------------------------------------------------
extern "C" void kernel_launch(void* const* d_in, const int* in_sizes, int n_in,
                              void* d_out, int out_size, void* d_ws, size_t ws_size,
                              hipStream_t stream) {
    const float* x  = (const float*)d_in[0];   // [8,64,128,128]
    const float* ow = (const float*)d_in[1];   // [18,64,3,3]
    const float* ob = (const float*)d_in[2];   // [18]
    const float* wt = (const float*)d_in[3];   // [64,64,3,3]
    float* out = (float*)d_out;                // [8,64,128,128]
    float* off = (float*)d_ws;                 // [8,18,128,128] scratch (9.4 MB)

    deform_offset_conv<<<(B_SZ * HW) / 256, 256, 0, stream>>>(x, ow, ob, off);
    deform_wmma_gemm<<<(B_SZ * HW) / 16, 128, 0, stream>>>(x, wt, off, out);
}